// SAGEConv_52046413693116
// MI455X (gfx1250) — compile-verified
//
#include <hip/hip_runtime.h>

#define D 128
#define A_STRIDE 260   // 256 floats + 4 pad -> conflict-free 16x4 fragment reads
#define W_STRIDE 130   // 128 float2 pairs + 2 pad (in float2 units)

typedef __attribute__((ext_vector_type(2))) float v2f;
typedef __attribute__((ext_vector_type(8))) float v8f;

// ---------------- zero workspace ----------------
__global__ void sage_zero_kernel(float4* __restrict__ p, int n4) {
    int i = blockIdx.x * blockDim.x + threadIdx.x;
    if (i < n4) {
        float4 z; z.x = 0.f; z.y = 0.f; z.z = 0.f; z.w = 0.f;
        p[i] = z;
    }
}

// ---------------- scatter: seg_sum += x[src], count[dst] += 1 ----------------
// one wave32 per edge: coalesced 512B row gather, 4 f32 atomics per lane
__global__ __launch_bounds__(256) void sage_scatter_kernel(
    const float* __restrict__ x, const int* __restrict__ edges,
    float* __restrict__ seg, float* __restrict__ cnt, int E) {
    const int lane = threadIdx.x & 31;
    const int e = blockIdx.x * 8 + (threadIdx.x >> 5);
    if (e >= E) return;
    const int s = edges[e];
    const int t = edges[E + e];
    const float4 v = *(const float4*)(x + (size_t)s * D + lane * 4);
    float* b = seg + (size_t)t * D + lane * 4;
    atomicAdd(b + 0, v.x);
    atomicAdd(b + 1, v.y);
    atomicAdd(b + 2, v.z);
    atomicAdd(b + 3, v.w);
    if (lane == 0) atomicAdd(cnt + t, 1.0f);
}

// ---------------- fused GEMM: out = [mean|x] @ [Wl;Ws]^T + (bl+bs) ----------------
// block = 256 threads = 8 waves; each wave computes a 16x128 output slab via
// v_wmma_f32_16x16x4_f32 over K=256, with double-buffered LDS fragment loads.
__global__ __launch_bounds__(256) void sage_gemm_kernel(
    const float* __restrict__ x, const float* __restrict__ seg,
    const float* __restrict__ cnt,
    const float* __restrict__ Wl, const float* __restrict__ bl,
    const float* __restrict__ Ws, const float* __restrict__ bs,
    float* __restrict__ out, int N) {

    extern __shared__ float smem[];
    float* lds_w   = smem;                          // 128 * W_STRIDE float2 = 33280 f
    float* lds_a   = smem + 128 * W_STRIDE * 2;     // 128 * A_STRIDE       = 33280 f
    float* lds_inv = lds_a + 128 * A_STRIDE;        // 128 f

    const int tid  = threadIdx.x;
    const int lane = tid & 31;
    const int wave = tid >> 5;
    const int m    = lane & 15;   // row of A-tile / col of B-tile
    const int half = lane >> 4;   // selects K pair within K=4 step
    const int rowbase = blockIdx.x * 128;

    // ---- stage fused weights, pair-interleaved, [n][p] layout ----
    for (int idx = tid; idx < 128 * 128; idx += 256) {
        const int n = idx >> 7;
        const int p = idx & 127;
        const int k = p * 2;
        v2f w;
        if (k < 128) w = *(const v2f*)(Wl + n * D + k);
        else         w = *(const v2f*)(Ws + n * D + (k - 128));
        *(v2f*)(lds_w + (n * W_STRIDE + p) * 2) = w;
    }
    // ---- 1 / max(count,1) per row ----
    for (int i = tid; i < 128; i += 256) {
        const int gr = rowbase + i;
        const float c = (gr < N) ? cnt[gr] : 1.0f;
        lds_inv[i] = 1.0f / fmaxf(c, 1.0f);
    }
    __syncthreads();
    // ---- stage A = [mean | x], coalesced float4 loads ----
    for (int idx = tid; idx < 128 * 64; idx += 256) {
        const int row = idx >> 6;
        const int c4  = (idx & 63) * 4;      // 0..252
        const int gr  = rowbase + row;
        float4 v; v.x = 0.f; v.y = 0.f; v.z = 0.f; v.w = 0.f;
        if (gr < N) {
            if (c4 < 128) {
                v = *(const float4*)(seg + (size_t)gr * D + c4);
                const float sc = lds_inv[row];
                v.x *= sc; v.y *= sc; v.z *= sc; v.w *= sc;
            } else {
                v = *(const float4*)(x + (size_t)gr * D + (c4 - 128));
            }
        }
        *(float4*)(lds_a + row * A_STRIDE + c4) = v;
    }
    __syncthreads();

    // ---- K=256 WMMA loop: 64 steps x 8 N-tiles, software pipelined ----
    const float* aRow = lds_a + (wave * 16 + m) * A_STRIDE;
    const float* wCol = lds_w + (m * W_STRIDE) * 2;   // + nt*16*W_STRIDE*2 per tile

    v8f acc[8];
#pragma unroll
    for (int nt = 0; nt < 8; ++nt) {
        const int n = nt * 16 + m;
        const float bias = bl[n] + bs[n];
        acc[nt] = (v8f){bias, bias, bias, bias, bias, bias, bias, bias};
    }

    // prologue: fragments for t = 0
    v2f a_cur = *(const v2f*)(aRow + half * 2);
    v2f b_cur[8];
#pragma unroll
    for (int nt = 0; nt < 8; ++nt)
        b_cur[nt] = *(const v2f*)(wCol + (nt * 16 * W_STRIDE + half) * 2);

#pragma unroll 4
    for (int t = 0; t < 64; ++t) {
        // issue next step's 9 fragment loads before consuming current ones
        const int tn = (t + 1) & 63;     // t=63 wraps (dead prefetch, in bounds)
        const int pn = tn * 2 + half;
        v2f a_nxt = *(const v2f*)(aRow + tn * 4 + half * 2);
        v2f b_nxt[8];
#pragma unroll
        for (int nt = 0; nt < 8; ++nt)
            b_nxt[nt] = *(const v2f*)(wCol + (nt * 16 * W_STRIDE + pn) * 2);

#pragma unroll
        for (int nt = 0; nt < 8; ++nt)
            acc[nt] = __builtin_amdgcn_wmma_f32_16x16x4_f32(
                false, a_cur, false, b_cur[nt], (short)0, acc[nt], false, false);

        a_cur = a_nxt;
#pragma unroll
        for (int nt = 0; nt < 8; ++nt) b_cur[nt] = b_nxt[nt];
    }

    // ---- transpose C through this wave's own (dead) A rows, store coalesced ----
    float* cBase = lds_a + (wave * 16) * A_STRIDE;
#pragma unroll
    for (int nt = 0; nt < 8; ++nt) {
#pragma unroll
        for (int vv = 0; vv < 8; ++vv) {
            cBase[(half * 8 + vv) * A_STRIDE + nt * 16 + m] = acc[nt][vv];
        }
    }
#pragma unroll
    for (int mm = 0; mm < 16; ++mm) {
        const int grow = rowbase + wave * 16 + mm;
        if (grow < N) {
            const float4 val = *(const float4*)(cBase + mm * A_STRIDE + lane * 4);
            *(float4*)(out + (size_t)grow * D + lane * 4) = val;
        }
    }
}

extern "C" void kernel_launch(void* const* d_in, const int* in_sizes, int n_in,
                              void* d_out, int out_size, void* d_ws, size_t ws_size,
                              hipStream_t stream) {
    const float* x     = (const float*)d_in[0];
    const int*   edges = (const int*)d_in[1];   // [2, E] int32 (JAX x64 off)
    const float* Wl    = (const float*)d_in[2];
    const float* bl    = (const float*)d_in[3];
    const float* Ws    = (const float*)d_in[4];
    const float* bs    = (const float*)d_in[5];
    float*       out   = (float*)d_out;

    const int N = in_sizes[0] / D;
    const int E = in_sizes[1] / 2;

    float* seg = (float*)d_ws;              // N*128 floats
    float* cnt = seg + (size_t)N * D;       // N floats

    const int n4 = (N * (D + 1) + 3) / 4;
    sage_zero_kernel<<<(n4 + 255) / 256, 256, 0, stream>>>((float4*)d_ws, n4);

    sage_scatter_kernel<<<(E + 7) / 8, 256, 0, stream>>>(x, edges, seg, cnt, E);

    const int smem_bytes = (128 * W_STRIDE * 2 + 128 * A_STRIDE + 128) * 4; // 266,752 B < 320KB WGP LDS
    hipFuncSetAttribute(reinterpret_cast<const void*>(sage_gemm_kernel),
                        hipFuncAttributeMaxDynamicSharedMemorySize, smem_bytes);
    sage_gemm_kernel<<<(N + 127) / 128, 256, smem_bytes, stream>>>(
        x, seg, cnt, Wl, bl, Ws, bs, out, N);
}